// A2SSM_2920577761491
// MI455X (gfx1250) — compile-verified
//
#include <hip/hip_runtime.h>
#include <hip/hip_bf16.h>

typedef __attribute__((ext_vector_type(2))) float v2f;
typedef __attribute__((ext_vector_type(8))) float v8f;

#define BB   16
#define CC   192
#define HH   64
#define WW   64
#define LL   (HH * WW)          // 4096
#define KPROJ 14                // dt_rank + 2*d_state
#define RANK  12                // dt_rank

// ---------------------------------------------------------------------------
// K1: depthwise 5x5 conv (SAME, cross-correlation) + bias + SiLU -> xs
// one block per (b,c) image; 68x68 halo tile in LDS
// ---------------------------------------------------------------------------
__global__ __launch_bounds__(256) void k_conv_silu(
    const float* __restrict__ x, const float* __restrict__ cw,
    const float* __restrict__ cb, float* __restrict__ xs)
{
    __shared__ float sm[68 * 68];
    const int blk = blockIdx.x;
    const int c = blk % CC;
    const int b = blk / CC;
    const float* xp = x + ((size_t)b * CC + c) * LL;

    for (int i = threadIdx.x; i < 68 * 68; i += 256) {
        int iy = i / 68 - 2, ix = i % 68 - 2;
        float v = 0.f;
        if ((unsigned)iy < (unsigned)HH && (unsigned)ix < (unsigned)WW)
            v = xp[iy * WW + ix];
        sm[i] = v;
    }
    float w[25];
#pragma unroll
    for (int t = 0; t < 25; ++t) w[t] = cw[c * 25 + t];
    const float bias = cb[c];
    __syncthreads();

    float* op = xs + ((size_t)b * CC + c) * LL;
    for (int p = threadIdx.x; p < LL; p += 256) {
        int h = p >> 6, wc = p & 63;
        float acc = bias;
#pragma unroll
        for (int kh = 0; kh < 5; ++kh)
#pragma unroll
            for (int kw = 0; kw < 5; ++kw)
                acc = fmaf(sm[(h + kh) * 68 + (wc + kw)], w[kh * 5 + kw], acc);
        op[p] = acc * (1.f / (1.f + __expf(-acc)));   // SiLU
    }
}

// ---------------------------------------------------------------------------
// K2: x_dbl[k,l] = sum_c x_proj_w[k,c] * xs[c,l]   (per batch)
// WMMA f32 16x16x4. 4 waves per block share one LDS weight copy; each wave
// owns a 16-wide l-tile. A fragment = contiguous float pair -> ds_load_b64.
// ---------------------------------------------------------------------------
__global__ __launch_bounds__(128) void k_xproj_wmma(
    const float* __restrict__ xs, const float* __restrict__ xw,
    float* __restrict__ xdbl)
{
    __shared__ float smw[16 * CC];        // zero-padded 16 x 192 weights
    for (int i = threadIdx.x; i < 16 * CC; i += 128) {
        int k = i / CC, cc = i % CC;
        smw[i] = (k < KPROJ) ? xw[k * CC + cc] : 0.f;
    }
    __syncthreads();

    const int lane = threadIdx.x & 31;
    const int wave = threadIdx.x >> 5;
    const int l0 = blockIdx.x * 64 + wave * 16;
    const int b  = blockIdx.y;
    const float* xsb = xs + (size_t)b * CC * LL;

    const int half = lane >> 4, lh = lane & 15;
    const int o0 = 2 * half;                  // ISA A-layout: V0=K0/K2, V1=K1/K3
    const int o1 = o0 + 1;

    v8f acc = {};
    for (int c0 = 0; c0 < CC; c0 += 4) {
        // contiguous 8B pair from LDS (single ds_load_b64)
        v2f a = *reinterpret_cast<const v2f*>(&smw[lh * CC + c0 + o0]);
        v2f bf;
        bf.x = xsb[(size_t)(c0 + o0) * LL + l0 + lh];
        bf.y = xsb[(size_t)(c0 + o1) * LL + l0 + lh];
        acc = __builtin_amdgcn_wmma_f32_16x16x4_f32(
                  false, a, false, bf, (short)0, acc, false, false);
    }
    float* o = xdbl + (size_t)b * KPROJ * LL + l0 + lh;
#pragma unroll
    for (int r = 0; r < 8; ++r) {
        int m = r + 8 * half;                 // C/D layout: lanes 16-31 hold M+8
        if (m < KPROJ) o[(size_t)m * LL] = acc[r];
    }
}

// ---------------------------------------------------------------------------
// K3: delta[c,l] = softplus(sum_r dt_w[c,r] * dts[r,l] + dt_b[c])
// WMMA f32 16x16x4, 3 K-steps (rank 12); 4 waves per block, one l-tile each.
// ---------------------------------------------------------------------------
__global__ __launch_bounds__(128) void k_delta_wmma(
    const float* __restrict__ xdbl, const float* __restrict__ dtw,
    const float* __restrict__ dtb, float* __restrict__ delta)
{
    const int lane = threadIdx.x & 31;
    const int wave = threadIdx.x >> 5;
    const int l0    = blockIdx.x * 64 + wave * 16;
    const int cbase = blockIdx.y * 16;
    const int b     = blockIdx.z;
    const float* dts = xdbl + (size_t)b * KPROJ * LL;   // rows 0..11 = dts

    const int half = lane >> 4, lh = lane & 15;
    const int o0 = 2 * half;
    const int o1 = o0 + 1;

    v8f acc = {};
#pragma unroll
    for (int k0 = 0; k0 < RANK; k0 += 4) {
        // contiguous 8B pair from global (single global_load_b64)
        v2f a = *reinterpret_cast<const v2f*>(&dtw[(cbase + lh) * RANK + k0 + o0]);
        v2f bf;
        bf.x = dts[(size_t)(k0 + o0) * LL + l0 + lh];
        bf.y = dts[(size_t)(k0 + o1) * LL + l0 + lh];
        acc = __builtin_amdgcn_wmma_f32_16x16x4_f32(
                  false, a, false, bf, (short)0, acc, false, false);
    }
#pragma unroll
    for (int r = 0; r < 8; ++r) {
        int c = cbase + r + 8 * half;
        float v = acc[r] + dtb[c];
        float sp = (v > 20.f) ? v : log1pf(__expf(v));  // softplus
        delta[((size_t)b * CC + c) * LL + l0 + lh] = sp;
    }
}

// ---------------------------------------------------------------------------
// K4: selective scan h_t = exp(delta*A)*h_{t-1} + delta*B_t*u_t, then
//     y = h*C_t + u*D.  One wave per (b,c); wave-parallel affine scan
//     (log2(32)=5 shuffle-compose steps), carry chained across 128 tiles.
//     Reads xs from d_out and overwrites with y in place.
// ---------------------------------------------------------------------------
__global__ __launch_bounds__(32) void k_scan(
    const float* __restrict__ xdbl, const float* __restrict__ delta_pre,
    const float* __restrict__ dtw,  const float* __restrict__ dtb,
    const float* __restrict__ A_logs, const float* __restrict__ Ds,
    float* __restrict__ y, int use_pre)
{
    const int lane = threadIdx.x;
    const int bc = blockIdx.x;
    const int c = bc % CC;
    const int b = bc / CC;

    float* yrow = y + (size_t)bc * LL;                  // xs lives here (in d_out)
    const float* xd   = xdbl + (size_t)b * KPROJ * LL;
    const float* Bs   = xd + (size_t)RANK * LL;         // row 12
    const float* Cs   = xd + (size_t)(RANK + 1) * LL;   // row 13
    const float* dpre = delta_pre + (size_t)bc * LL;

    const float Ac = -__expf(A_logs[c]);                // d_state = 1
    const float Dc = Ds[c];
    const float dtbias = dtb[c];
    float dw[RANK];
    if (!use_pre) {
#pragma unroll
        for (int r = 0; r < RANK; ++r) dw[r] = dtw[c * RANK + r];
    }

    float hcarry = 0.f;
    for (int t0 = 0; t0 < LL; t0 += 32) {
        const int l = t0 + lane;
        const float u  = yrow[l];
        const float Bv = Bs[l];
        const float Cv = Cs[l];

        float d;
        if (use_pre) {
            d = dpre[l];
        } else {
            float v = dtbias;
#pragma unroll
            for (int r = 0; r < RANK; ++r)
                v = fmaf(dw[r], xd[(size_t)r * LL + l], v);
            d = (v > 20.f) ? v : log1pf(__expf(v));
        }

        float a  = __expf(d * Ac);
        float bb = d * Bv * u;

        // inclusive scan of affine maps across the wave (wave32)
#pragma unroll
        for (int off = 1; off < 32; off <<= 1) {
            float a_up = __shfl_up(a, (unsigned)off, 32);
            float b_up = __shfl_up(bb, (unsigned)off, 32);
            if (lane >= off) { bb = fmaf(a, b_up, bb); a *= a_up; }
        }
        const float h = fmaf(a, hcarry, bb);
        yrow[l] = fmaf(h, Cv, u * Dc);
        hcarry = __shfl(h, 31, 32);                     // carry to next tile
    }
}

// ---------------------------------------------------------------------------
extern "C" void kernel_launch(void* const* d_in, const int* in_sizes, int n_in,
                              void* d_out, int out_size, void* d_ws, size_t ws_size,
                              hipStream_t stream)
{
    const float* x        = (const float*)d_in[0];
    const float* conv_w   = (const float*)d_in[1];
    const float* conv_b   = (const float*)d_in[2];
    const float* x_proj_w = (const float*)d_in[3];
    const float* dt_w     = (const float*)d_in[4];
    const float* dt_b     = (const float*)d_in[5];
    const float* A_logs   = (const float*)d_in[6];
    const float* Ds       = (const float*)d_in[7];
    float* out = (float*)d_out;
    float* ws  = (float*)d_ws;

    const size_t xdbl_elems  = (size_t)BB * KPROJ * LL;          // 3.7 MB
    const size_t delta_elems = (size_t)BB * CC * LL;             // 50.3 MB
    float* xdbl  = ws;
    float* delta = ws + xdbl_elems;
    const int use_pre =
        (ws_size >= (xdbl_elems + delta_elems) * sizeof(float)) ? 1 : 0;

    // K1: conv + SiLU, xs staged in d_out (same size as final output)
    k_conv_silu<<<BB * CC, 256, 0, stream>>>(x, conv_w, conv_b, out);

    // K2: x-projection GEMM (WMMA f32 16x16x4), 4 waves/block share weights
    dim3 g2(LL / 64, BB);
    k_xproj_wmma<<<g2, 128, 0, stream>>>(out, x_proj_w, xdbl);

    // K3: dt-projection + softplus (WMMA), if workspace allows
    if (use_pre) {
        dim3 g3(LL / 64, CC / 16, BB);
        k_delta_wmma<<<g3, 128, 0, stream>>>(xdbl, dt_w, dt_b, delta);
    }

    // K4: wave-parallel selective scan + output, in-place over d_out
    k_scan<<<BB * CC, 32, 0, stream>>>(xdbl, delta, dt_w, dt_b,
                                       A_logs, Ds, out, use_pre);
}